// TtMixtralAttention_21388937134653
// MI455X (gfx1250) — compile-verified
//
#include <hip/hip_runtime.h>
#include <hip/hip_bf16.h>

// ---------------------------------------------------------------------------
// Mixtral decode attention (N_DEV=8, B=32, H=4096, D=128, NQ=4, NKV=1, W=4096)
// fp32 end-to-end, all GEMMs on V_WMMA_F32_16X16X4_F32 (wave32).
// Memory-bound: KV cache (2 x 537 MB) read exactly once.
// ---------------------------------------------------------------------------

typedef float v2f __attribute__((ext_vector_type(2)));
typedef float v8f __attribute__((ext_vector_type(8)));

#define NDEV 8
#define BSZ  32
#define HDIM 4096
#define DHEAD 128
#define NQ   4
#define QKV  768      // (4+2)*128
#define WIN  4096

__device__ __forceinline__ v8f wmma_f32(v2f a, v2f b, v8f c) {
  // D = A(16x4,f32) * B(4x16,f32) + C(16x16,f32)
  return __builtin_amdgcn_wmma_f32_16x16x4_f32(
      /*neg_a=*/false, a, /*neg_b=*/false, b,
      /*c_mod=*/(short)0, c, /*reuse_a=*/false, /*reuse_b=*/false);
}

// ---------------------------------------------------------------------------
// K1: xqkv[n,b,q] = sum_h x[b,h] * wqkv[n,h,q]   (per device: 32x4096 @ 4096x768)
// one wave per (dev, 16-col tile); wave computes both 16-row tiles (b=0..31).
// ---------------------------------------------------------------------------
__global__ void k_qkv(const float* __restrict__ x, const float* __restrict__ wqkv,
                      float* __restrict__ xqkv) {
  int wave = (blockIdx.x * blockDim.x + threadIdx.x) >> 5;
  int lane = threadIdx.x & 31;
  int n  = wave / 48;
  int jt = wave % 48;                // 48 tiles of 16 cols = 768
  const float* W = wqkv + (size_t)n * HDIM * QKV;
  int l = lane & 15, lh = lane >> 4;
  v8f c0 = {}; v8f c1 = {};
  for (int k0 = 0; k0 < HDIM; k0 += 4) {
    int ka = k0 + 2 * lh;
    v2f a0, a1, b;
    a0.x = x[(size_t)l * HDIM + ka];
    a0.y = x[(size_t)l * HDIM + ka + 1];
    a1.x = x[(size_t)(l + 16) * HDIM + ka];
    a1.y = x[(size_t)(l + 16) * HDIM + ka + 1];
    b.x  = W[(size_t)ka * QKV + jt * 16 + l];
    b.y  = W[(size_t)(ka + 1) * QKV + jt * 16 + l];
    c0 = wmma_f32(a0, b, c0);
    c1 = wmma_f32(a1, b, c1);
  }
  int col = jt * 16 + l;
#pragma unroll
  for (int j = 0; j < 8; ++j) {
    int m = j + 8 * lh;
    xqkv[((size_t)n * BSZ + m) * QKV + col]        = c0[j];
    xqkv[((size_t)n * BSZ + 16 + m) * QKV + col]   = c1[j];
  }
}

// ---------------------------------------------------------------------------
// K2: q_rot = (q @ rot) * D^-1/2 ; k_rot = k @ rot
// per device: 160 rows (128 q rows + 32 k rows) x 128 @ 128x128.
// wave = (n, mtile 0..9, ntile 0..7). Tiles 0..7 are q, 8..9 are k.
// ---------------------------------------------------------------------------
__global__ void k_rot(const float* __restrict__ xqkv, const float* __restrict__ rot,
                      float* __restrict__ qrot, float* __restrict__ krot) {
  int wave = (blockIdx.x * blockDim.x + threadIdx.x) >> 5;
  int lane = threadIdx.x & 31;
  int n   = wave / 80;
  int rem = wave % 80;
  int mt  = rem / 8;
  int nt  = rem % 8;
  int l = lane & 15, lh = lane >> 4;
  int m = mt * 16 + l;               // row 0..159
  const float* arow;
  if (m < 128) {                      // q row: b = m>>2, h = m&3
    arow = xqkv + ((size_t)n * BSZ + (m >> 2)) * QKV + (m & 3) * DHEAD;
  } else {                            // k row: b = m-128
    arow = xqkv + ((size_t)n * BSZ + (m - 128)) * QKV + NQ * DHEAD;
  }
  v8f c = {};
  for (int k0 = 0; k0 < DHEAD; k0 += 4) {
    int ka = k0 + 2 * lh;
    v2f a, b;
    a.x = arow[ka];
    a.y = arow[ka + 1];
    b.x = rot[(size_t)ka * DHEAD + nt * 16 + l];
    b.y = rot[(size_t)(ka + 1) * DHEAD + nt * 16 + l];
    c = wmma_f32(a, b, c);
  }
  const float scale = 0.08838834764831845f;   // 128^-0.5
  int e = nt * 16 + l;
#pragma unroll
  for (int j = 0; j < 8; ++j) {
    int mo = mt * 16 + j + 8 * lh;
    if (mo < 128) {
      qrot[((size_t)n * 128 + mo) * DHEAD + e] = c[j] * scale;  // (n*32+b)*4+h rows
    } else {
      krot[((size_t)n * BSZ + (mo - 128)) * DHEAD + e] = c[j];
    }
  }
}

// ---------------------------------------------------------------------------
// K3: scores[nb, h, p] = q_rot[nb,h,:] . key[nb,p,:]
// key row p = (p==cur) ? k_rot : cache_k. -inf past layer_slice.
// wave = (nb 0..255, ptile 0..255). A padded 4->16 rows with zeros.
// Each key row (512 B) streamed exactly once.
// ---------------------------------------------------------------------------
__global__ void k_scores(const float* __restrict__ qrot, const float* __restrict__ krot,
                         const float* __restrict__ cache_k,
                         const int* __restrict__ start_pos_p,
                         const int* __restrict__ cur_pos_p,
                         float* __restrict__ scores) {
  int wave = (blockIdx.x * blockDim.x + threadIdx.x) >> 5;
  int lane = threadIdx.x & 31;
  int pt = wave & 255;
  int nb = wave >> 8;                // n*32+b
  int cur = *cur_pos_p;
  int layer_slice = min(*start_pos_p + 1, WIN);
  int l = lane & 15, lh = lane >> 4;
  const float* Q  = qrot + (size_t)nb * NQ * DHEAD;
  int p = pt * 16 + l;               // this lane's key row
  const float* krow = (p == cur) ? (krot + (size_t)nb * DHEAD)
                                 : (cache_k + ((size_t)nb * WIN + p) * DHEAD);
  const float* qrow = Q + (size_t)((l < NQ) ? l : 0) * DHEAD;   // in-bounds clamp
  float amask = (l < NQ) ? 1.0f : 0.0f;
  v8f c = {};
  for (int k0 = 0; k0 < DHEAD; k0 += 4) {
    int ka = k0 + 2 * lh;
    v2f a, b;
    a.x = qrow[ka] * amask;
    a.y = qrow[ka + 1] * amask;
    b.x = krow[ka];
    b.y = krow[ka + 1];
    c = wmma_f32(a, b, c);
  }
  if (lh == 0) {                      // rows 0..3 live in lanes 0..15, vgpr j=0..3
    int pcol = pt * 16 + l;
    float pad = -__builtin_inff();
#pragma unroll
    for (int j = 0; j < NQ; ++j) {
      float val = (pcol < layer_slice) ? c[j] : pad;
      scores[((size_t)nb * NQ + j) * WIN + pcol] = val;
    }
  }
}

// ---------------------------------------------------------------------------
// K4: numerically stable softmax over each 4096-wide row (1024 rows).
// ---------------------------------------------------------------------------
__global__ void k_softmax(float* __restrict__ s_all) {
  float* s = s_all + (size_t)blockIdx.x * WIN;
  int tid = threadIdx.x;              // 256 threads, 16 elems each
  float v[16];
  float m = -__builtin_inff();
#pragma unroll
  for (int i = 0; i < 16; ++i) {
    v[i] = s[tid + i * 256];
    m = fmaxf(m, v[i]);
  }
#pragma unroll
  for (int off = 16; off > 0; off >>= 1)
    m = fmaxf(m, __shfl_xor(m, off, 32));
  __shared__ float sm[8];
  __shared__ float ss[8];
  int wid = tid >> 5;
  if ((tid & 31) == 0) sm[wid] = m;
  __syncthreads();
  float bm = -__builtin_inff();
#pragma unroll
  for (int i = 0; i < 8; ++i) bm = fmaxf(bm, sm[i]);
  float sum = 0.0f;
#pragma unroll
  for (int i = 0; i < 16; ++i) {
    v[i] = expf(v[i] - bm);
    sum += v[i];
  }
#pragma unroll
  for (int off = 16; off > 0; off >>= 1)
    sum += __shfl_xor(sum, off, 32);
  if ((tid & 31) == 0) ss[wid] = sum;
  __syncthreads();
  float bs = 0.0f;
#pragma unroll
  for (int i = 0; i < 8; ++i) bs += ss[i];
  float inv = 1.0f / bs;
#pragma unroll
  for (int i = 0; i < 16; ++i)
    s[tid + i * 256] = v[i] * inv;
}

// ---------------------------------------------------------------------------
// K5: out[nb, h, d] = sum_p attn[nb,h,p] * val[nb,p,d]
// val row p = (p==cur) ? v_new (xqkv slice) : cache_v.
// wave = (nb 0..255, kchunk 0..15): 256-position K slice, 8 N-tile accumulators,
// f32 atomicAdd into zero-initialized ws. cache_v streamed exactly once.
// ---------------------------------------------------------------------------
__global__ void k_av(const float* __restrict__ attn, const float* __restrict__ cache_v,
                     const float* __restrict__ xqkv, const int* __restrict__ cur_pos_p,
                     float* __restrict__ out) {
  int wave = (blockIdx.x * blockDim.x + threadIdx.x) >> 5;
  int lane = threadIdx.x & 31;
  int chunk = wave & 15;
  int nb    = wave >> 4;
  int cur = *cur_pos_p;
  int l = lane & 15, lh = lane >> 4;
  const float* A    = attn + (size_t)nb * NQ * WIN;
  const float* V    = cache_v + (size_t)nb * WIN * DHEAD;
  const float* vnew = xqkv + (size_t)nb * QKV + (NQ + 1) * DHEAD;
  const float* arow = A + (size_t)((l < NQ) ? l : 0) * WIN;
  float amask = (l < NQ) ? 1.0f : 0.0f;
  v8f c[8];
#pragma unroll
  for (int i = 0; i < 8; ++i) c[i] = (v8f){};
  int pbase = chunk * 256;
  for (int k0 = 0; k0 < 256; k0 += 4) {
    int pa = pbase + k0 + 2 * lh;     // this half-wave's two K rows
    v2f a;
    a.x = arow[pa] * amask;
    a.y = arow[pa + 1] * amask;
    const float* vr0 = (pa == cur)     ? vnew : V + (size_t)pa * DHEAD;
    const float* vr1 = (pa + 1 == cur) ? vnew : V + (size_t)(pa + 1) * DHEAD;
#pragma unroll
    for (int nt = 0; nt < 8; ++nt) {
      v2f b;
      b.x = vr0[nt * 16 + l];
      b.y = vr1[nt * 16 + l];
      c[nt] = wmma_f32(a, b, c[nt]);
    }
  }
  if (lh == 0) {
#pragma unroll
    for (int nt = 0; nt < 8; ++nt)
#pragma unroll
      for (int j = 0; j < NQ; ++j)
        atomicAdd(&out[(size_t)nb * (NQ * DHEAD) + j * DHEAD + nt * 16 + l], c[nt][j]);
  }
}

// ---------------------------------------------------------------------------
// K6: y[b,h] = sum_n sum_e out[n,b,e] * wo[n,e,h]   (32x512 @ 512x4096, 8 devs)
// wave = ntile (0..255), two M-tile accumulators, device sum folded into K loop.
// ---------------------------------------------------------------------------
__global__ void k_dense(const float* __restrict__ out, const float* __restrict__ wo,
                        float* __restrict__ y) {
  int wave = (blockIdx.x * blockDim.x + threadIdx.x) >> 5;
  int lane = threadIdx.x & 31;
  int nt = wave;
  int l = lane & 15, lh = lane >> 4;
  v8f c0 = {}; v8f c1 = {};
  for (int n = 0; n < NDEV; ++n) {
    const float* A  = out + (size_t)n * BSZ * (NQ * DHEAD);
    const float* Wn = wo + (size_t)n * (NQ * DHEAD) * HDIM;
    for (int k0 = 0; k0 < NQ * DHEAD; k0 += 4) {
      int ka = k0 + 2 * lh;
      v2f a0, a1, b;
      a0.x = A[(size_t)l * (NQ * DHEAD) + ka];
      a0.y = A[(size_t)l * (NQ * DHEAD) + ka + 1];
      a1.x = A[(size_t)(l + 16) * (NQ * DHEAD) + ka];
      a1.y = A[(size_t)(l + 16) * (NQ * DHEAD) + ka + 1];
      b.x  = Wn[(size_t)ka * HDIM + nt * 16 + l];
      b.y  = Wn[(size_t)(ka + 1) * HDIM + nt * 16 + l];
      c0 = wmma_f32(a0, b, c0);
      c1 = wmma_f32(a1, b, c1);
    }
  }
  int col = nt * 16 + l;
#pragma unroll
  for (int j = 0; j < 8; ++j) {
    int m = j + 8 * lh;
    y[(size_t)m * HDIM + col]        = c0[j];
    y[(size_t)(16 + m) * HDIM + col] = c1[j];
  }
}

// ---------------------------------------------------------------------------
extern "C" void kernel_launch(void* const* d_in, const int* in_sizes, int n_in,
                              void* d_out, int out_size, void* d_ws, size_t ws_size,
                              hipStream_t stream) {
  (void)in_sizes; (void)n_in; (void)out_size; (void)ws_size;
  const float* x       = (const float*)d_in[0];
  const float* wqkv    = (const float*)d_in[1];
  const float* wo      = (const float*)d_in[2];
  const float* cache_k = (const float*)d_in[3];
  const float* cache_v = (const float*)d_in[4];
  const float* rot     = (const float*)d_in[5];
  const int* start_pos = (const int*)d_in[6];
  const int* cur_pos   = (const int*)d_in[7];

  // workspace layout (floats)
  float* ws      = (float*)d_ws;
  float* xqkv_ws = ws;                                   // 8*32*768   = 196608
  float* qrot    = xqkv_ws + NDEV * BSZ * QKV;           // 8*32*4*128 = 131072
  float* krot    = qrot + NDEV * BSZ * NQ * DHEAD;       // 8*32*128   = 32768
  float* scores  = krot + NDEV * BSZ * DHEAD;            // 8*32*4*4096= 4194304
  float* outw    = scores + (size_t)NDEV * BSZ * NQ * WIN; // 8*32*512 = 131072

  hipMemsetAsync(outw, 0, (size_t)NDEV * BSZ * NQ * DHEAD * sizeof(float), stream);

  k_qkv    <<<48,   256, 0, stream>>>(x, wqkv, xqkv_ws);                 // 384 waves
  k_rot    <<<80,   256, 0, stream>>>(xqkv_ws, rot, qrot, krot);         // 640 waves
  k_scores <<<8192, 256, 0, stream>>>(qrot, krot, cache_k, start_pos,
                                      cur_pos, scores);                  // 65536 waves
  k_softmax<<<1024, 256, 0, stream>>>(scores);
  k_av     <<<512,  256, 0, stream>>>(scores, cache_v, xqkv_ws, cur_pos, outw); // 4096 waves
  k_dense  <<<32,   256, 0, stream>>>(outw, wo, (float*)d_out);          // 256 waves
}